// ALNet_85126251807339
// MI455X (gfx1250) — compile-verified
//
#include <hip/hip_runtime.h>

typedef __attribute__((ext_vector_type(2))) float v2f;
typedef __attribute__((ext_vector_type(4))) float v4f;
typedef __attribute__((ext_vector_type(8))) float v8f;

#define F_DIM 256
#define P_LEAVES 128
#define WAVES_PER_WG 8
#define ROWS_PER_WAVE 32      // two 16-row M tiles per wave
#define THREADS_PER_WG 256

// Conflict-free LDS layout for B fragments:
// word(n,k) = (k>>2)*SLICE + ((k>>1)&1)*KH_OFF + n*2 + (k&1)
// KH_OFF=288 == 32 (mod 64) -> lanes 0-15 (K 0,1) and 16-31 (K 2,3) hit
// complementary 32-bank halves; all 64 banks distinct per ds_load_b64.
#define SLICE_W 576           // words per kc slice (128 n * 4 k + pad)
#define KH_OFF  288           // words between K-half planes (== 32 mod 64)
#define LDS_WORDS ((F_DIM / 4) * SLICE_W)   // 36864 words = 147456 B

// min/max via v_med3_f32 (raw intrinsic: no canonicalize ops inserted,
// exact min/max for the finite values this network produces).
__device__ __forceinline__ float fmin1(float a, float b) {
    return __builtin_amdgcn_fmed3f(a, b, -__builtin_inff());
}
__device__ __forceinline__ float fmax1(float a, float b) {
    return __builtin_amdgcn_fmed3f(a, b, __builtin_inff());
}
__device__ __forceinline__ v8f v8min(v8f a, v8f b) {
    v8f r;
#pragma unroll
    for (int i = 0; i < 8; ++i) r[i] = fmin1(a[i], b[i]);
    return r;
}
__device__ __forceinline__ v8f v8max(v8f a, v8f b) {
    v8f r;
#pragma unroll
    for (int i = 0; i < 8; ++i) r[i] = fmax1(a[i], b[i]);
    return r;
}
// Single-instruction xor-butterfly via ds_swizzle_b32 (group-of-32:
// offset = (xor_mask<<10) | (or_mask<<5) | and_mask, and_mask=0x1f).
template <int XORM>
__device__ __forceinline__ v8f v8swz(v8f v) {
    constexpr int PAT = (XORM << 10) | 0x1f;
    v8f r;
#pragma unroll
    for (int i = 0; i < 8; ++i)
        r[i] = __int_as_float(
            __builtin_amdgcn_ds_swizzle(__float_as_int(v[i]), PAT));
    return r;
}

extern "C" __global__ __launch_bounds__(THREADS_PER_WG)
void aln_wmma_kernel(const float* __restrict__ x,
                     const float* __restrict__ W,
                     const float* __restrict__ bias,
                     float* __restrict__ out)
{
    extern __shared__ float ldsW[];

    const int tid = threadIdx.x;

    // ---- Stage W (k-major [256][128]) into the khalf-split LDS layout ----
    const v4f* W4 = (const v4f*)W;
#pragma unroll 4
    for (int i = 0; i < (F_DIM * P_LEAVES) / (4 * THREADS_PER_WG); ++i) {
        int flat4 = tid + i * THREADS_PER_WG;
        v4f w = W4[flat4];
        int k  = flat4 >> 5;            // flat/128
        int n0 = (flat4 & 31) * 4;      // flat%128
        int bse = (k >> 2) * SLICE_W + ((k >> 1) & 1) * KH_OFF + (k & 1);
        ldsW[bse + (n0 + 0) * 2] = w[0];
        ldsW[bse + (n0 + 1) * 2] = w[1];
        ldsW[bse + (n0 + 2) * 2] = w[2];
        ldsW[bse + (n0 + 3) * 2] = w[3];
    }

    const int lane  = tid & 31;
    const int wave  = tid >> 5;
    const int c     = lane & 15;   // N-idx (B/C/D frags) and M-idx (A frag)
    const int khalf = lane >> 4;   // K-half select for A/B fragments

    const int rowBase = (blockIdx.x * WAVES_PER_WG + wave) * ROWS_PER_WAVE;

    // Bias per lane: leaf n = nt*16 + c (same for all 8 M components)
    float bb[8];
#pragma unroll
    for (int nt = 0; nt < 8; ++nt) bb[nt] = bias[nt * 16 + c];

    __syncthreads();

    // A-fragment sources for the two M tiles (streamed once -> non-temporal)
    const float* xA0 = x + (size_t)(rowBase + c) * F_DIM + khalf * 2;
    const float* xA1 = xA0 + 16 * F_DIM;

    // Lane-constant part of the B-fragment LDS address
    const float* ldsB = ldsW + khalf * KH_OFF + c * 2;

    v8f acc[2][8];
#pragma unroll
    for (int mt = 0; mt < 2; ++mt)
#pragma unroll
        for (int nt = 0; nt < 8; ++nt) acc[mt][nt] = (v8f)0.0f;

    // ---- GEMM: K=256 in chunks of 4, each B fragment feeds 2 WMMAs ----
#pragma unroll 2
    for (int kc = 0; kc < F_DIM / 4; ++kc) {
        v2f a0 = __builtin_nontemporal_load((const v2f*)(xA0 + kc * 4));
        v2f a1 = __builtin_nontemporal_load((const v2f*)(xA1 + kc * 4));
        const float* bp = ldsB + kc * SLICE_W;
#pragma unroll
        for (int nt = 0; nt < 8; ++nt) {
            v2f bf = *(const v2f*)(bp + nt * 32);        // ds_load_b64, 0-conflict
            acc[0][nt] = __builtin_amdgcn_wmma_f32_16x16x4_f32(
                false, a0, false, bf, (short)0, acc[0][nt], false, false);
            acc[1][nt] = __builtin_amdgcn_wmma_f32_16x16x4_f32(
                false, a1, false, bf, (short)0, acc[1][nt], false, false);
        }
    }

    // ---- Bias + min/max tree fold over 128 leaves per M tile ----
    // Level order (deepest->root): min(c0), max(c1), min(c2), max(c3),
    //                              min(nt0), max(nt1), min(nt2).
    // After the xor-1 min, lane pairs hold duplicates, so pack tile 2j into
    // even lanes and tile 2j+1 into odd lanes; stages xor2/4/8 keep the two
    // streams separate (partners share lane bit0), and the nt-bit0 min
    // becomes a final xor-1 fold.
    const bool oddlane = (lane & 1) != 0;
#pragma unroll
    for (int mt = 0; mt < 2; ++mt) {
        v8f p[4];
#pragma unroll
        for (int j = 0; j < 4; ++j) {
            v8f va = acc[mt][2 * j]     + bb[2 * j];
            v8f vb = acc[mt][2 * j + 1] + bb[2 * j + 1];
            va = v8min(va, v8swz<1>(va));            // min over c bit0
            vb = v8min(vb, v8swz<1>(vb));
            v8f m;
#pragma unroll
            for (int i = 0; i < 8; ++i) m[i] = oddlane ? vb[i] : va[i];
            m = v8max(m, v8swz<2>(m));               // max over c bit1
            m = v8min(m, v8swz<4>(m));               // min over c bit2
            m = v8max(m, v8swz<8>(m));               // max over c bit3
            m = v8min(m, v8swz<1>(m));               // min over nt bit0
            p[j] = m;
        }
        v8f q0 = v8max(p[0], p[1]);                  // max over nt bit1
        v8f q1 = v8max(p[2], p[3]);
        v8f r  = v8min(q0, q1);                      // min over nt bit2

        // lanes 0-15 carry M=0..7, lanes 16-31 carry M=8..15
        if (c == 0) {
            float* o = out + rowBase + mt * 16 + khalf * 8;
            v4f lo = { r[0], r[1], r[2], r[3] };
            v4f hi = { r[4], r[5], r[6], r[7] };
            *(v4f*)(o + 0) = lo;                     // global_store_b128
            *(v4f*)(o + 4) = hi;
        }
    }
}

extern "C" void kernel_launch(void* const* d_in, const int* in_sizes, int n_in,
                              void* d_out, int out_size, void* d_ws, size_t ws_size,
                              hipStream_t stream) {
    const float* x = (const float*)d_in[0];   // [B, 256] f32
    const float* W = (const float*)d_in[1];   // [256, 128] f32
    const float* b = (const float*)d_in[2];   // [128] f32
    float* out = (float*)d_out;               // [B] f32

    const int Bsz = in_sizes[0] / F_DIM;                      // 65536
    const int rowsPerWG = WAVES_PER_WG * ROWS_PER_WAVE;       // 256
    const int nBlocks = Bsz / rowsPerWG;                      // 256

    const size_t shmem = (size_t)LDS_WORDS * sizeof(float);   // 147456 B
    (void)hipFuncSetAttribute((const void*)aln_wmma_kernel,
                              hipFuncAttributeMaxDynamicSharedMemorySize,
                              (int)shmem);

    aln_wmma_kernel<<<nBlocks, THREADS_PER_WG, shmem, stream>>>(x, W, b, out);
}